// TabularDecoder_84482006712637
// MI455X (gfx1250) — compile-verified
//
#include <hip/hip_runtime.h>

typedef __attribute__((ext_vector_type(2))) float v2f;
typedef __attribute__((ext_vector_type(4))) float v4f;
typedef __attribute__((ext_vector_type(8))) float v8f;

#define B_ROWS 16384
#define NNUM   32
#define NCAT   8
#define NCOLS  40
#define DD     256
#define CC     16
#define APAD   260   // 256 dwords + 4 pad -> bank-conflict-free A fragment reads

// Block = 128 threads (4 waves). blockIdx.x in [0,256): group of 4 row-tiles.
// blockIdx.y in [0,40): table column (decoder selector is uniform per block).
__global__ __launch_bounds__(128) void tabdec_wmma_kernel(
    const float* __restrict__ emb,
    const float* __restrict__ w_num,  const float* __restrict__ b_num,
    const float* __restrict__ W_cat,  const float* __restrict__ b_cat,
    const float* __restrict__ w_miss, const float* __restrict__ b_miss,
    float* __restrict__ out_num, float* __restrict__ out_cat,
    float* __restrict__ out_miss)
{
    // B matrices in K-paired layout: element (k, c) at [(k>>1)*32 + c*2 + (k&1)]
    __shared__ float Bls [(DD / 2) * 32];   // 16 KB: main decoder weights
    __shared__ float B2ls[(DD / 2) * 32];   // 16 KB: miss decoder (cat columns)
    __shared__ float Als [4][16 * APAD];    // 4 waves x (16 rows x 260) ~66.6 KB

    const int tid  = threadIdx.x;
    const int wave = tid >> 5;
    const int lane = tid & 31;
    const int sel  = (lane >> 4) & 1;   // which K-pair of the 4-wide K slab
    const int hn   = lane & 15;         // A: row M; B/D: column N
    const int j    = blockIdx.y;
    const bool isCat = (j >= NNUM);
    const int b0   = (blockIdx.x * 4 + wave) * 16;   // first cell row of tile

    // ---- Stage A tile: 16 rows x 256 f32, coalesced float4 loads ----
    {
        const float* abase = emb + ((size_t)b0 * NCOLS + j) * DD;
        float* As = &Als[wave][0];
#pragma unroll
        for (int it = 0; it < 32; ++it) {
            int t  = it * 32 + lane;     // float4 index in tile [0,1024)
            int m  = t >> 6;             // row within tile
            int c4 = t & 63;             // float4 within row
            v4f v = *(const v4f*)(abase + (size_t)m * NCOLS * DD + c4 * 4);
            *(v4f*)(As + m * APAD + c4 * 4) = v;
        }
    }

    // ---- Stage B: phase 1 (copy / zero) ----
    if (isCat) {
        const float* wsrc = W_cat + (size_t)(j - NNUM) * DD * CC;
        for (int t = tid; t < DD * CC; t += 128) {
            int k = t >> 4, c = t & 15;
            Bls[(k >> 1) * 32 + c * 2 + (k & 1)] = wsrc[t];
        }
        for (int t = tid; t < DD * CC; t += 128) B2ls[t] = 0.0f;
    } else {
        for (int t = tid; t < DD * CC; t += 128) Bls[t] = 0.0f;
    }
    __syncthreads();

    // ---- Stage B: phase 2 (inject w_num / w_miss columns) ----
    if (isCat) {
        for (int k = tid; k < DD; k += 128)
            B2ls[(k >> 1) * 32 + (k & 1)] = w_miss[k];           // column 0
    } else {
        for (int k = tid; k < DD; k += 128) {
            Bls[(k >> 1) * 32 + 0 * 2 + (k & 1)] = w_num[k];     // column 0
            Bls[(k >> 1) * 32 + 1 * 2 + (k & 1)] = w_miss[k];    // column 1
        }
    }
    __syncthreads();

    // ---- K loop: D=256 in steps of 4 with v_wmma_f32_16x16x4_f32 ----
    v8f acc0 = {};
    v8f acc1 = {};
    const float* Arow = &Als[wave][hn * APAD];

    if (isCat) {
#pragma unroll 4
        for (int k = 0; k < DD; k += 4) {
            v2f a  = *(const v2f*)(Arow + k + sel * 2);
            v2f bb = *(const v2f*)(Bls  + ((k >> 1) + sel) * 32 + hn * 2);
            v2f bm = *(const v2f*)(B2ls + ((k >> 1) + sel) * 32 + hn * 2);
            acc0 = __builtin_amdgcn_wmma_f32_16x16x4_f32(
                       false, a, false, bb, (short)0, acc0, false, false);
            acc1 = __builtin_amdgcn_wmma_f32_16x16x4_f32(
                       false, a, false, bm, (short)0, acc1, false, false);
        }
    } else {
#pragma unroll 4
        for (int k = 0; k < DD; k += 4) {
            v2f a  = *(const v2f*)(Arow + k + sel * 2);
            v2f bb = *(const v2f*)(Bls  + ((k >> 1) + sel) * 32 + hn * 2);
            acc0 = __builtin_amdgcn_wmma_f32_16x16x4_f32(
                       false, a, false, bb, (short)0, acc0, false, false);
        }
    }

    // ---- Epilogue: D layout -> lane holds D[M = r + 8*sel][N = hn] ----
    if (isCat) {
        const int jc = j - NNUM;
        const float bc = b_cat[jc * CC + hn];
        const float bm = b_miss[0];
#pragma unroll
        for (int r = 0; r < 8; ++r) {
            int row = b0 + r + 8 * sel;
            out_cat[(size_t)row * (NCAT * CC) + jc * CC + hn] = acc0[r] + bc;
        }
        if (hn == 0) {
#pragma unroll
            for (int r = 0; r < 8; ++r) {
                int row = b0 + r + 8 * sel;
                out_miss[(size_t)row * NCOLS + j] = acc1[r] + bm;
            }
        }
    } else {
        const float bn = b_num[0];
        const float bm = b_miss[0];
        if (hn == 0) {
#pragma unroll
            for (int r = 0; r < 8; ++r) {
                int row = b0 + r + 8 * sel;
                out_num[(size_t)row * NNUM + j] = acc0[r] + bn;
            }
        } else if (hn == 1) {
#pragma unroll
            for (int r = 0; r < 8; ++r) {
                int row = b0 + r + 8 * sel;
                out_miss[(size_t)row * NCOLS + j] = acc0[r] + bm;
            }
        }
    }
}

extern "C" void kernel_launch(void* const* d_in, const int* in_sizes, int n_in,
                              void* d_out, int out_size, void* d_ws, size_t ws_size,
                              hipStream_t stream) {
    (void)in_sizes; (void)n_in; (void)out_size; (void)d_ws; (void)ws_size;
    const float* emb    = (const float*)d_in[0];
    const float* w_num  = (const float*)d_in[1];
    const float* b_num  = (const float*)d_in[2];
    const float* W_cat  = (const float*)d_in[3];
    const float* b_cat  = (const float*)d_in[4];
    const float* w_miss = (const float*)d_in[5];
    const float* b_miss = (const float*)d_in[6];

    float* out      = (float*)d_out;
    float* out_num  = out;                                  // [16384, 32]
    float* out_cat  = out_num + (size_t)B_ROWS * NNUM;      // [16384, 8, 16]
    float* out_miss = out_cat + (size_t)B_ROWS * NCAT * CC; // [16384, 40]

    dim3 grid(B_ROWS / 64, NCOLS);   // 256 x 40 blocks, 128 threads each
    tabdec_wmma_kernel<<<grid, 128, 0, stream>>>(
        emb, w_num, b_num, W_cat, b_cat, w_miss, b_miss,
        out_num, out_cat, out_miss);
}